// ACPNet_50044958933188
// MI455X (gfx1250) — compile-verified
//
#include <hip/hip_runtime.h>
#include <math.h>

#define NB 8
#define NPTS 2048
#define KNB 20

typedef __attribute__((ext_vector_type(2))) float v2f;
typedef __attribute__((ext_vector_type(8))) float v8f;

__device__ __forceinline__ v8f wmma_f32_16x16x4(v2f a, v2f b, v8f c) {
  // (neg_a, A, neg_b, B, c_mod, C, reuse_a, reuse_b)
  return __builtin_amdgcn_wmma_f32_16x16x4_f32(false, a, false, b, (short)0, c, false, false);
}

// ---------------------------------------------------------------------------
// WMMA KNN: X is [B][Kp][NPTS] with zero-padded channel rows (Kp % 4 == 0).
// Block = 256 threads (8 waves) owns 256 consecutive rows. Wave w owns two
// 16-row tiles. A fragments (X^T row block) preloaded in registers; sweep all
// 128 column tiles with f32 WMMA, spill dot tiles to LDS, scan threads keep
// per-row top-20 of d = 2*dot - |x_n|^2 - |x_m|^2 via a branchless,
// statically-indexed predicated insert (arrays stay in VGPRs).
// ---------------------------------------------------------------------------
template <int Kp>
__global__ void knn_wmma_kernel(const float* __restrict__ X, int* __restrict__ idx) {
  constexpr int KI = Kp / 4;
  __shared__ float dts[256][20];  // dot tiles; 80B row stride keeps 16B align
  __shared__ float txx[NPTS];     // column squared norms
  const int b = blockIdx.y;
  const int row0 = blockIdx.x * 256;
  const float* Xb = X + (size_t)b * Kp * NPTS;
  const int tid = threadIdx.x;
  const int wave = tid >> 5, lane = tid & 31;
  const int l16 = lane & 15, lh = lane >> 4;

  // column squared norms for the whole point set
  for (int m = tid; m < NPTS; m += 256) {
    float s = 0.f;
#pragma unroll
    for (int c = 0; c < Kp; ++c) { float v = Xb[(size_t)c * NPTS + m]; s += v * v; }
    txx[m] = s;
  }
  // own-row squared norm
  const int myrow = row0 + tid;
  float xx = 0.f;
#pragma unroll
  for (int c = 0; c < Kp; ++c) { float v = Xb[(size_t)c * NPTS + myrow]; xx += v * v; }

  // preload A fragments for this wave's two row tiles
  v2f a0[KI], a1[KI];
  const int rb0 = row0 + wave * 32;
  const int rb1 = rb0 + 16;
#pragma unroll
  for (int i = 0; i < KI; ++i) {
    int ka = i * 4 + lh * 2;
    a0[i].x = Xb[(size_t)ka * NPTS + rb0 + l16];
    a0[i].y = Xb[(size_t)(ka + 1) * NPTS + rb0 + l16];
    a1[i].x = Xb[(size_t)ka * NPTS + rb1 + l16];
    a1[i].y = Xb[(size_t)(ka + 1) * NPTS + rb1 + l16];
  }

  float bestv[KNB];
  int besti[KNB];
#pragma unroll
  for (int i = 0; i < KNB; ++i) { bestv[i] = -3.0e38f; besti[i] = 0; }

  __syncthreads();  // txx ready

  for (int j = 0; j < NPTS / 16; ++j) {
    const int cb = j * 16;
    v8f acc0 = (v8f){0.f, 0.f, 0.f, 0.f, 0.f, 0.f, 0.f, 0.f};
    v8f acc1 = (v8f){0.f, 0.f, 0.f, 0.f, 0.f, 0.f, 0.f, 0.f};
#pragma unroll
    for (int i = 0; i < KI; ++i) {
      int ka = i * 4 + lh * 2;
      v2f bf;
      bf.x = Xb[(size_t)ka * NPTS + cb + l16];
      bf.y = Xb[(size_t)(ka + 1) * NPTS + cb + l16];
      acc0 = wmma_f32_16x16x4(a0[i], bf, acc0);
      acc1 = wmma_f32_16x16x4(a1[i], bf, acc1);
    }
    __syncthreads();  // previous scan done before overwriting dts
#pragma unroll
    for (int r = 0; r < 8; ++r) {
      dts[wave * 32 + r + lh * 8][l16] = acc0[r];
      dts[wave * 32 + 16 + r + lh * 8][l16] = acc1[r];
    }
    __syncthreads();
    // scan: thread owns local row `tid`
    const float* drow = dts[tid];
#pragma unroll
    for (int q = 0; q < 16; ++q) {
      float d = 2.f * drow[q] - xx - txx[cb + q];
      if (d > bestv[KNB - 1]) {
        const int mi = cb + q;
        // fully unrolled predicated insert: static indices only -> stays in VGPRs
#pragma unroll
        for (int p = KNB - 1; p >= 1; --p) {
          bool up = d > bestv[p - 1];            // insertion point above p: shift
          bool here = (!up) && (d > bestv[p]);   // insert exactly at slot p
          float pv = bestv[p - 1];
          int pi = besti[p - 1];
          bestv[p] = up ? pv : (here ? d : bestv[p]);
          besti[p] = up ? pi : (here ? mi : besti[p]);
        }
        if (d > bestv[0]) { bestv[0] = d; besti[0] = mi; }
      }
    }
  }
  int* o = idx + ((size_t)b * NPTS + myrow) * KNB;
#pragma unroll
  for (int i = 0; i < KNB; ++i) o[i] = besti[i];
}

// ---------------------------------------------------------------------------
// Pad x [B,3,N] -> xpad [B,4,N] (row 3 zero)
// ---------------------------------------------------------------------------
__global__ void pad_x_kernel(const float* __restrict__ x, float* __restrict__ xp) {
  size_t t = (size_t)blockIdx.x * blockDim.x + threadIdx.x;
  if (t >= (size_t)NB * 4 * NPTS) return;
  int n = (int)(t % NPTS);
  int r = (int)((t / NPTS) % 4);
  int b = (int)(t / (4 * NPTS));
  xp[t] = (r < 3) ? x[(size_t)b * 3 * NPTS + (size_t)r * NPTS + n] : 0.f;
}

// ---------------------------------------------------------------------------
// LAFE -> x_manet [B,20,N]: rows 0..2 = x, rows 3..18 = elu(pooled edge feat),
// row 19 = 0 (K padding)
// ---------------------------------------------------------------------------
__global__ void lafe_kernel(const float* __restrict__ x, const int* __restrict__ idx0,
                            const float* __restrict__ w1, const float* __restrict__ b1,
                            const float* __restrict__ w2, const float* __restrict__ b2,
                            const float* __restrict__ w3, const float* __restrict__ b3,
                            float* __restrict__ xman) {
  const int t = blockIdx.x * blockDim.x + threadIdx.x;
  const int b = t / NPTS, n = t % NPTS;
  const float* xb = x + (size_t)b * 3 * NPTS;
  const float xc0 = xb[n], xc1 = xb[NPTS + n], xc2 = xb[2 * NPTS + n];

  float sa = b3[0];
#pragma unroll
  for (int o = 0; o < 16; ++o) {
    float nf = w1[o * 3] * xc0 + w1[o * 3 + 1] * xc1 + w1[o * 3 + 2] * xc2 + b1[o];
    sa += w3[o] * nf;
  }

  const int* id = idx0 + ((size_t)b * NPTS + n) * KNB;
  float logit[KNB];
#pragma unroll
  for (int k = 0; k < KNB; ++k) {
    int m = id[k];
    float d0 = xc0 - xb[m], d1 = xc1 - xb[NPTS + m], d2 = xc2 - xb[2 * NPTS + m];
    float na = b3[0];
#pragma unroll
    for (int o = 0; o < 16; ++o) {
      float ef = w2[o * 3] * d0 + w2[o * 3 + 1] * d1 + w2[o * 3 + 2] * d2 + b2[o];
      na += w3[o] * ef;
    }
    float l = sa + na;
    logit[k] = l > 0.f ? l : 0.01f * l;  // leaky_relu(.., 0.01)
  }
  float mx = -3.0e38f;
#pragma unroll
  for (int k = 0; k < KNB; ++k) mx = fmaxf(mx, logit[k]);
  float sum = 0.f;
#pragma unroll
  for (int k = 0; k < KNB; ++k) { logit[k] = expf(logit[k] - mx); sum += logit[k]; }
  float inv = 1.f / sum;

  float vals[16];
#pragma unroll
  for (int o = 0; o < 16; ++o) vals[o] = 0.f;
#pragma unroll
  for (int k = 0; k < KNB; ++k) {
    int m = id[k];
    float d0 = xc0 - xb[m], d1 = xc1 - xb[NPTS + m], d2 = xc2 - xb[2 * NPTS + m];
    float coef = logit[k] * inv;
#pragma unroll
    for (int o = 0; o < 16; ++o) {
      float ef = w2[o * 3] * d0 + w2[o * 3 + 1] * d1 + w2[o * 3 + 2] * d2 + b2[o];
      vals[o] += coef * ef;
    }
  }
  float* xm = xman + (size_t)b * 20 * NPTS;
  xm[n] = xc0; xm[NPTS + n] = xc1; xm[2 * NPTS + n] = xc2;
#pragma unroll
  for (int o = 0; o < 16; ++o) {
    float v = vals[o];
    v = v > 0.f ? v : (expf(v) - 1.f);  // elu
    xm[(3 + o) * NPTS + n] = v;
  }
  xm[19 * NPTS + n] = 0.f;  // K-pad row
}

// ---------------------------------------------------------------------------
// EdgeConv weight split with K padding:
//   wN[o,c] = w[o,c] - w[o,2C+c] ; wC[o,c] = w[o,C+c] + w[o,2C+c] ; pad -> 0
// ---------------------------------------------------------------------------
__global__ void combine_w_kernel(const float* __restrict__ w, float* __restrict__ wN,
                                 float* __restrict__ wC, int O, int Cc, int Kp) {
  int t = blockIdx.x * blockDim.x + threadIdx.x;
  if (t >= O * Kp) return;
  int o = t / Kp, c = t % Kp;
  float vn = 0.f, vc = 0.f;
  if (c < Cc) {
    float a = w[o * 3 * Cc + c];
    float bb = w[o * 3 * Cc + Cc + c];
    float d = w[o * 3 * Cc + 2 * Cc + c];
    vn = a - d;
    vc = bb + d;
  }
  wN[t] = vn;
  wC[t] = vc;
}

// pack [M,K] -> zero-padded [M,Kp]
__global__ void pack_pad_kernel(const float* __restrict__ src, float* __restrict__ dst,
                                int M, int K, int Kp) {
  int t = blockIdx.x * blockDim.x + threadIdx.x;
  if (t >= M * Kp) return;
  int m = t / Kp, c = t % Kp;
  dst[t] = (c < K) ? src[m * K + c] : 0.f;
}

// zero one channel row across batches; dst pre-offset to the row start
__global__ void zero_row_kernel(float* __restrict__ dst, int batchStride) {
  int t = blockIdx.x * blockDim.x + threadIdx.x;
  if (t >= NB * NPTS) return;
  int b = t / NPTS, n = t % NPTS;
  dst[(size_t)b * batchStride + n] = 0.f;
}

// ---------------------------------------------------------------------------
// WMMA f32 GEMM: out[b] = act(bn(W @ Xin[b] + bias)), KD compile-time, %4==0.
// W: [M,KD] row-major (padded); Xin: [B][KD][Ncols]; out: [B][M][Ncols]
// ---------------------------------------------------------------------------
template <int KD>
__global__ void gemm_wmma_kernel(const float* __restrict__ W, const float* __restrict__ Xin,
                                 const float* __restrict__ bias, const float* __restrict__ bn,
                                 float* __restrict__ out, int M, int Ncols, int act,
                                 float slope, int xBatchStride, int outBatchStride) {
  const int b = blockIdx.z;
  const int wave = threadIdx.x >> 5;
  const int lane = threadIdx.x & 31;
  const int mt = blockIdx.y * 8 + wave;
  if (mt * 16 >= M) return;
  const int n0 = blockIdx.x * 16;
  const int m0 = mt * 16;
  const int l16 = lane & 15;
  const int lh = lane >> 4;
  const float* Xb = Xin + (size_t)b * xBatchStride;
  float* Ob = out + (size_t)b * outBatchStride;
  const float* Wr = W + (size_t)(m0 + l16) * KD + lh * 2;

  v8f acc = (v8f){0.f, 0.f, 0.f, 0.f, 0.f, 0.f, 0.f, 0.f};
#pragma unroll 4
  for (int kk = 0; kk < KD; kk += 4) {
    int ka = kk + lh * 2;
    v2f a = *(const v2f*)(Wr + kk);
    v2f bf;
    bf.x = Xb[(size_t)ka * Ncols + n0 + l16];
    bf.y = Xb[(size_t)(ka + 1) * Ncols + n0 + l16];
    acc = wmma_f32_16x16x4(a, bf, acc);
  }
#pragma unroll
  for (int r = 0; r < 8; ++r) {
    int m = m0 + r + lh * 8;
    float v = acc[r];
    if (bias) v += bias[m];
    if (bn) {
      float g = bn[m], be = bn[M + m], mu = bn[2 * M + m], var = bn[3 * M + m];
      v = (v - mu) * (g * rsqrtf(var + 1e-5f)) + be;
    }
    if (act == 1) v = fmaxf(v, 0.f);
    else if (act == 2) v = v > 0.f ? v : slope * v;
    Ob[(size_t)m * Ncols + n0 + l16] = v;
  }
}

// ---------------------------------------------------------------------------
// Fused attention pool (one block per point):
//   x[c,k] = leaky0.2(bnConv(HN[c,idx[n,k]] + HC[c,n]))
//   h[c,k] = relu(bnP(x[c,k]))
//   att    = p_w @ h + p_b  (WMMA; cols padded via zeroed clamp column)
//   out[c] = sum_k x[c,k]*softmax_k(att[c,k])
// ---------------------------------------------------------------------------
template <int C>
__global__ void att_pool_kernel(const float* __restrict__ HN, const float* __restrict__ HC,
                                const int* __restrict__ idx, const float* __restrict__ bnConv,
                                const float* __restrict__ bnP, const float* __restrict__ pw,
                                const float* __restrict__ pb, float* __restrict__ outp,
                                int outBatchStride) {
  __shared__ float xt[C][KNB];
  __shared__ float ht[C][KNB + 1];  // extra zero column for padded WMMA cols
  __shared__ float at[C][KNB];
  const int b = blockIdx.y;
  const int n = blockIdx.x;
  const int c = threadIdx.x;
  const float* HNb = HN + (size_t)b * C * NPTS;
  const float* HCb = HC + (size_t)b * C * NPTS;
  const int* id = idx + ((size_t)b * NPTS + n) * KNB;

  const float s1 = bnConv[c] * rsqrtf(bnConv[3 * C + c] + 1e-5f);
  const float mu1 = bnConv[2 * C + c], be1 = bnConv[C + c];
  const float s2 = bnP[c] * rsqrtf(bnP[3 * C + c] + 1e-5f);
  const float mu2 = bnP[2 * C + c], be2 = bnP[C + c];
  const float hc = HCb[(size_t)c * NPTS + n];

#pragma unroll
  for (int k = 0; k < KNB; ++k) {
    int m = id[k];
    float v = HNb[(size_t)c * NPTS + m] + hc;
    v = (v - mu1) * s1 + be1;
    v = v > 0.f ? v : 0.2f * v;  // leaky 0.2 after conv bn
    xt[c][k] = v;
    float h = (v - mu2) * s2 + be2;
    ht[c][k] = fmaxf(h, 0.f);
  }
  ht[c][KNB] = 0.f;
  __syncthreads();

  const int wave = threadIdx.x >> 5;
  const int lane = threadIdx.x & 31;
  const int l16 = lane & 15;
  const int lh = lane >> 4;
  constexpr int NW = C / 32;
  constexpr int TOT = (C / 16) * 2;  // row tiles x 2 col tiles (20 -> 32 cols)
  for (int tile = wave; tile < TOT; tile += NW) {
    int mt = tile >> 1, nt = tile & 1;
    int m0 = mt * 16, nn0 = nt * 16;
    int col = nn0 + l16;
    int colc = col < KNB ? col : KNB;  // clamp to zero column
    const float* pwr = pw + (size_t)(m0 + l16) * C + lh * 2;
    v8f acc = (v8f){0.f, 0.f, 0.f, 0.f, 0.f, 0.f, 0.f, 0.f};
#pragma unroll 4
    for (int kk = 0; kk < C; kk += 4) {
      int ka = kk + lh * 2;
      v2f a = *(const v2f*)(pwr + kk);
      v2f bf;
      bf.x = ht[ka][colc];
      bf.y = ht[ka + 1][colc];
      acc = wmma_f32_16x16x4(a, bf, acc);
    }
#pragma unroll
    for (int r = 0; r < 8; ++r) {
      int m = m0 + r + lh * 8;
      if (col < KNB) at[m][col] = acc[r] + pb[m];
    }
  }
  __syncthreads();

  float mx = -3.0e38f;
#pragma unroll
  for (int k = 0; k < KNB; ++k) mx = fmaxf(mx, at[c][k]);
  float e[KNB];
  float sum = 0.f;
#pragma unroll
  for (int k = 0; k < KNB; ++k) { e[k] = expf(at[c][k] - mx); sum += e[k]; }
  float o = 0.f;
  float inv = 1.f / sum;
#pragma unroll
  for (int k = 0; k < KNB; ++k) o += xt[c][k] * (e[k] * inv);
  outp[(size_t)b * outBatchStride + (size_t)c * NPTS + n] = o;
}

// ---------------------------------------------------------------------------
__global__ void copy_rows_kernel(const float* __restrict__ src, float* __restrict__ dst,
                                 int rows, int srcBS, int dstBS) {
  size_t t = (size_t)blockIdx.x * blockDim.x + threadIdx.x;
  size_t tot = (size_t)NB * rows * NPTS;
  if (t >= tot) return;
  int n = (int)(t % NPTS);
  int r = (int)((t / NPTS) % rows);
  int b = (int)(t / ((size_t)NPTS * rows));
  dst[(size_t)b * dstBS + (size_t)r * NPTS + n] = src[(size_t)b * srcBS + (size_t)r * NPTS + n];
}

// ---------------------------------------------------------------------------
extern "C" void kernel_launch(void* const* d_in, const int* in_sizes, int n_in,
                              void* d_out, int out_size, void* d_ws, size_t ws_size,
                              hipStream_t stream) {
  (void)in_sizes; (void)n_in; (void)out_size; (void)ws_size;
  const float* x = (const float*)d_in[0];
  const float* la_w1 = (const float*)d_in[1];
  const float* la_b1 = (const float*)d_in[2];
  const float* la_w2 = (const float*)d_in[3];
  const float* la_b2 = (const float*)d_in[4];
  const float* la_w3 = (const float*)d_in[5];
  const float* la_b3 = (const float*)d_in[6];
  const float* mlp_w = (const float*)d_in[7];
  const float* mlp_b = (const float*)d_in[8];
  const float* mlp_bn = (const float*)d_in[9];
  const float* w1 = (const float*)d_in[10];
  const float* bn1 = (const float*)d_in[11];
  const float* w2 = (const float*)d_in[12];
  const float* bn2 = (const float*)d_in[13];
  const float* w3 = (const float*)d_in[14];
  const float* bn3 = (const float*)d_in[15];
  const float* p1_bn = (const float*)d_in[16];
  const float* p1_w = (const float*)d_in[17];
  const float* p1_b = (const float*)d_in[18];
  const float* p2_bn = (const float*)d_in[19];
  const float* p2_w = (const float*)d_in[20];
  const float* p2_b = (const float*)d_in[21];
  float* out = (float*)d_out;

  // --- bump allocator on workspace ---
  char* ws = (char*)d_ws;
  size_t off = 0;
  auto alloc = [&](size_t bytes) -> void* {
    void* p = ws + off;
    off = (off + bytes + 255) & ~(size_t)255;
    return p;
  };
  float* xpad = (float*)alloc(sizeof(float) * NB * 4 * NPTS);      // x padded to 4 ch
  float* x_manet = (float*)alloc(sizeof(float) * NB * 20 * NPTS);  // 19 ch + pad
  int* idx0 = (int*)alloc(sizeof(int) * NB * NPTS * KNB);
  int* idx1 = (int*)alloc(sizeof(int) * NB * NPTS * KNB);
  int* idx2 = (int*)alloc(sizeof(int) * NB * NPTS * KNB);
  float* x_mlp = (float*)alloc(sizeof(float) * NB * 256 * NPTS);
  float* HN1 = (float*)alloc(sizeof(float) * NB * 64 * NPTS);
  float* HC1 = (float*)alloc(sizeof(float) * NB * 64 * NPTS);
  float* xc1 = (float*)alloc(sizeof(float) * NB * 84 * NPTS);      // 83 ch + pad
  float* HN2 = (float*)alloc(sizeof(float) * NB * 256 * NPTS);
  float* HC2 = (float*)alloc(sizeof(float) * NB * 256 * NPTS);
  float* xc2 = (float*)alloc(sizeof(float) * NB * 512 * NPTS);
  float* wN1 = (float*)alloc(sizeof(float) * 64 * 20);
  float* wC1 = (float*)alloc(sizeof(float) * 64 * 20);
  float* wN2 = (float*)alloc(sizeof(float) * 256 * 84);
  float* wC2 = (float*)alloc(sizeof(float) * 256 * 84);
  float* mlp_wp = (float*)alloc(sizeof(float) * 256 * 20);

  // 1) KNN on raw x (pad to 4 channels, Gram via WMMA)
  pad_x_kernel<<<(NB * 4 * NPTS + 255) / 256, 256, 0, stream>>>(x, xpad);
  knn_wmma_kernel<4><<<dim3(NPTS / 256, NB), 256, 0, stream>>>(xpad, idx0);
  // 2) LAFE -> x_manet [B,20,N] (row 19 zeroed)
  lafe_kernel<<<(NB * NPTS) / 256, 256, 0, stream>>>(x, idx0, la_w1, la_b1, la_w2, la_b2,
                                                     la_w3, la_b3, x_manet);
  // 3) weight prep (padded)
  combine_w_kernel<<<(64 * 20 + 255) / 256, 256, 0, stream>>>(w1, wN1, wC1, 64, 19, 20);
  combine_w_kernel<<<(256 * 84 + 255) / 256, 256, 0, stream>>>(w2, wN2, wC2, 256, 83, 84);
  pack_pad_kernel<<<(256 * 20 + 255) / 256, 256, 0, stream>>>(mlp_w, mlp_wp, 256, 19, 20);
  // 4) x_mlp = relu(bn(mlp_w @ x_manet + b))
  gemm_wmma_kernel<20><<<dim3(NPTS / 16, 2, NB), 256, 0, stream>>>(
      mlp_wp, x_manet, mlp_b, mlp_bn, x_mlp, 256, NPTS, 1, 0.f, 20 * NPTS, 256 * NPTS);
  // 5) KNN on x_manet (19 ch)
  knn_wmma_kernel<20><<<dim3(NPTS / 256, NB), 256, 0, stream>>>(x_manet, idx1);
  // 6) HN1/HC1 = wN1/wC1 @ x_manet (raw, pre-bn)
  gemm_wmma_kernel<20><<<dim3(NPTS / 16, 1, NB), 256, 0, stream>>>(
      wN1, x_manet, nullptr, nullptr, HN1, 64, NPTS, 0, 0.f, 20 * NPTS, 64 * NPTS);
  gemm_wmma_kernel<20><<<dim3(NPTS / 16, 1, NB), 256, 0, stream>>>(
      wC1, x_manet, nullptr, nullptr, HC1, 64, NPTS, 0, 0.f, 20 * NPTS, 64 * NPTS);
  // 7) attention pool 1 -> xc1 rows 19..82 ; copy rows 0..18 ; zero pad row 83
  att_pool_kernel<64><<<dim3(NPTS, NB), 64, 0, stream>>>(HN1, HC1, idx1, bn1, p1_bn, p1_w,
                                                         p1_b, xc1 + 19 * NPTS, 84 * NPTS);
  copy_rows_kernel<<<((size_t)NB * 19 * NPTS + 255) / 256, 256, 0, stream>>>(
      x_manet, xc1, 19, 20 * NPTS, 84 * NPTS);
  zero_row_kernel<<<(NB * NPTS + 255) / 256, 256, 0, stream>>>(xc1 + 83 * NPTS, 84 * NPTS);
  // 8) KNN on x_c1 (83 ch)
  knn_wmma_kernel<84><<<dim3(NPTS / 256, NB), 256, 0, stream>>>(xc1, idx2);
  // 9) HN2/HC2 = wN2/wC2 @ x_c1
  gemm_wmma_kernel<84><<<dim3(NPTS / 16, 2, NB), 256, 0, stream>>>(
      wN2, xc1, nullptr, nullptr, HN2, 256, NPTS, 0, 0.f, 84 * NPTS, 256 * NPTS);
  gemm_wmma_kernel<84><<<dim3(NPTS / 16, 2, NB), 256, 0, stream>>>(
      wC2, xc1, nullptr, nullptr, HC2, 256, NPTS, 0, 0.f, 84 * NPTS, 256 * NPTS);
  // 10) attention pool 2 -> xc2 rows 256..511 ; copy x_mlp -> rows 0..255
  att_pool_kernel<256><<<dim3(NPTS, NB), 256, 0, stream>>>(HN2, HC2, idx2, bn2, p2_bn, p2_w,
                                                           p2_b, xc2 + 256 * NPTS, 512 * NPTS);
  copy_rows_kernel<<<((size_t)NB * 256 * NPTS + 255) / 256, 256, 0, stream>>>(
      x_mlp, xc2, 256, 256 * NPTS, 512 * NPTS);
  // 11) out = leaky0.2(bn3(w3 @ x_c2))  (Kd=512, no padding needed)
  gemm_wmma_kernel<512><<<dim3(NPTS / 16, 4, NB), 256, 0, stream>>>(
      w3, xc2, nullptr, bn3, out, 512, NPTS, 2, 0.2f, 512 * NPTS, 512 * NPTS);
}